// CrossModalAlignment_83932250898702
// MI455X (gfx1250) — compile-verified
//
#include <hip/hip_runtime.h>

typedef __attribute__((ext_vector_type(2))) float v2f;
typedef __attribute__((ext_vector_type(8))) float v8f;

#define WMMA_F32_4(a, b, c) \
  __builtin_amdgcn_wmma_f32_16x16x4_f32(false, (a), false, (b), (short)0, (c), false, false)

__device__ __forceinline__ float wave_sum32(float v) {
#pragma unroll
  for (int off = 16; off > 0; off >>= 1) v += __shfl_xor(v, off, 32);
  return v;
}

// Gram G = X * X^T for X (16 x 256, row-major) using V_WMMA_F32_16X16X4_F32.
// A-fragment == B-fragment for a gram: lane L supplies X[L&15][k + 2*(L>>4) + {0,1}].
__device__ __forceinline__ v8f gram16_256(const float* __restrict__ X, int lane) {
  const int r15 = lane & 15;
  const int half = lane >> 4;
  const float* base = X + r15 * 256 + 2 * half;
  v8f c = {};
#pragma unroll 4
  for (int k = 0; k < 256; k += 4) {
    const v2f a = *(const v2f*)(base + k);
    c = WMMA_F32_4(a, a, c);
  }
  return c;
}

// Gauss-Jordan inverse of 16x16 SPD matrix. sA holds (G + lam*I) on entry and
// is destroyed; sV holds I on entry and inv on exit. Single-wave workgroup.
__device__ void gj_invert16(float (*sA)[17], float (*sV)[17], float* sF, int lane) {
  const int c = lane & 15;
  for (int p = 0; p < 16; ++p) {
    __syncthreads();
    const float ip = 1.0f / sA[p][p];
    __syncthreads();
    if (lane < 16) sA[p][c] *= ip; else sV[p][c] *= ip;
    __syncthreads();
    if (lane < 16) sF[lane] = (lane == p) ? 0.0f : sA[lane][p];
    __syncthreads();
#pragma unroll
    for (int t = 0; t < 16; ++t) {
      const int idx = t * 32 + lane;     // 0..511: [0,256)=A, [256,512)=V
      const int rr = (idx >> 4) & 15;
      const int cc = idx & 15;
      if (rr != p) {
        const float f = sF[rr];
        if (idx < 256) sA[rr][cc] -= f * sA[p][cc];
        else           sV[rr][cc] -= f * sV[p][cc];
      }
    }
  }
  __syncthreads();
}

// Per-image-batch: G_I[b] and S1[b] = (rho*P - I)^2 (16x16 each) into workspace.
__global__ void prep_img_kernel(const float* __restrict__ img,
                                const float* __restrict__ r,
                                float* __restrict__ wsGI,
                                float* __restrict__ wsS1,
                                float Bf) {
  __shared__ float Gm[16][16];
  __shared__ float sA[16][17];
  __shared__ float sV[16][17];
  __shared__ float sF[16];
  __shared__ float Mm[16][16];

  const int b = blockIdx.x;
  const int lane = threadIdx.x;
  const int l15 = lane & 15;
  const int half = lane >> 4;

  const float lam = (Bf / 256.0f) * expf(r[0]) + 1e-6f;
  const float rho = expf(r[1]);

  const v8f g = gram16_256(img + (size_t)b * (16 * 256), lane);
#pragma unroll
  for (int v = 0; v < 8; ++v) {
    const int i = v + 8 * half;          // C/D layout: M = v + 8*half, N = l15
    Gm[i][l15] = g[v];
    wsGI[b * 256 + i * 16 + l15] = g[v];
  }
  __syncthreads();

  for (int t = lane; t < 256; t += 32) {
    const int i = t >> 4, j = t & 15;
    sA[i][j] = Gm[i][j] + ((i == j) ? lam : 0.0f);
    sV[i][j] = (i == j) ? 1.0f : 0.0f;
  }
  gj_invert16(sA, sV, sF, lane);

  // M = (rho-1)*I - rho*lam*inv(G+lam I)   (symmetric)
  for (int t = lane; t < 256; t += 32) {
    const int i = t >> 4, j = t & 15;
    Mm[i][j] = ((i == j) ? (rho - 1.0f) : 0.0f) - rho * lam * sV[i][j];
  }
  __syncthreads();

  // S1 = M*M via WMMA (A-frag == B-frag since M symmetric)
  v8f s = {};
#pragma unroll
  for (int k = 0; k < 16; k += 4) {
    v2f a;
    a.x = Mm[l15][k + 2 * half];
    a.y = Mm[l15][k + 2 * half + 1];
    s = WMMA_F32_4(a, a, s);
  }
#pragma unroll
  for (int v = 0; v < 8; ++v) {
    const int i = v + 8 * half;
    wsS1[b * 256 + i * 16 + l15] = s[v];
  }
}

// Per-text: gram, inverse, S2 = M^2, then rc[b][n] for b=0..3 (B fixed at 4).
__global__ void per_text_kernel(const float* __restrict__ txt,
                                const float* __restrict__ r,
                                const float* __restrict__ alp_p,
                                const float* __restrict__ scale_p,
                                const float* __restrict__ wsGI,
                                const float* __restrict__ wsS1,
                                float* __restrict__ logits,
                                int N, float Nf) {
  __shared__ float Gm[16][16];
  __shared__ float sA[16][17];
  __shared__ float sV[16][17];
  __shared__ float sF[16];
  __shared__ float Mm[16][16];

  const int n = blockIdx.x;
  const int lane = threadIdx.x;
  const int l15 = lane & 15;
  const int half = lane >> 4;

  const float lam = (Nf / 256.0f) * expf(r[2]) + 1e-6f;
  const float rho = expf(r[3]);
  const float alp = alp_p[0];
  const float scl = scale_p[0];

  const v8f g = gram16_256(txt + (size_t)n * (16 * 256), lane);
#pragma unroll
  for (int v = 0; v < 8; ++v) Gm[v + 8 * half][l15] = g[v];
  __syncthreads();

  for (int t = lane; t < 256; t += 32) {
    const int i = t >> 4, j = t & 15;
    sA[i][j] = Gm[i][j] + ((i == j) ? lam : 0.0f);
    sV[i][j] = (i == j) ? 1.0f : 0.0f;
  }
  gj_invert16(sA, sV, sF, lane);

  for (int t = lane; t < 256; t += 32) {
    const int i = t >> 4, j = t & 15;
    Mm[i][j] = ((i == j) ? (rho - 1.0f) : 0.0f) - rho * lam * sV[i][j];
  }
  __syncthreads();

  v8f s = {};
#pragma unroll
  for (int k = 0; k < 16; k += 4) {
    v2f a;
    a.x = Mm[l15][k + 2 * half];
    a.y = Mm[l15][k + 2 * half + 1];
    s = WMMA_F32_4(a, a, s);
  }

  // rc[b][n] = -(alp/16)*<S1[b], G_T[n]> - ((1-alp)/16)*<S2[n], G_I[b]>
  float res[4];
#pragma unroll
  for (int b = 0; b < 4; ++b) {
    float p1 = 0.0f, p2 = 0.0f;
#pragma unroll
    for (int v = 0; v < 8; ++v) {
      const int i = v + 8 * half;
      p1 += wsS1[b * 256 + i * 16 + l15] * Gm[i][l15];
      p2 += s[v] * wsGI[b * 256 + i * 16 + l15];
    }
    const float partial = -(alp * 0.0625f) * p1 - ((1.0f - alp) * 0.0625f) * p2;
    res[b] = wave_sum32(partial);
  }
  if (lane == 0) {
#pragma unroll
    for (int b = 0; b < 4; ++b) logits[(size_t)b * N + n] = res[b] * scl;
  }
}

// In-place log_softmax over each row of out[B][N]; also writes sigmoid scalar.
__global__ void row_log_softmax_kernel(float* __restrict__ out,
                                       const float* __restrict__ lsc,
                                       int N, int B) {
  __shared__ float red[256];
  const int b = blockIdx.x;
  const int tid = threadIdx.x;
  float* row = out + (size_t)b * N;

  float mx = -3.402823466e+38f;
  for (int i = tid; i < N; i += 256) mx = fmaxf(mx, row[i]);
  red[tid] = mx; __syncthreads();
  for (int sft = 128; sft > 0; sft >>= 1) {
    if (tid < sft) red[tid] = fmaxf(red[tid], red[tid + sft]);
    __syncthreads();
  }
  mx = red[0];
  __syncthreads();

  float se = 0.0f;
  for (int i = tid; i < N; i += 256) se += expf(row[i] - mx);
  red[tid] = se; __syncthreads();
  for (int sft = 128; sft > 0; sft >>= 1) {
    if (tid < sft) red[tid] += red[tid + sft];
    __syncthreads();
  }
  const float lse = mx + logf(red[0]);
  __syncthreads();

  for (int i = tid; i < N; i += 256) row[i] = row[i] - lse;

  if (b == 0 && tid == 0) out[(size_t)B * N] = 1.0f / (1.0f + expf(-lsc[0]));
}

extern "C" void kernel_launch(void* const* d_in, const int* in_sizes, int n_in,
                              void* d_out, int out_size, void* d_ws, size_t ws_size,
                              hipStream_t stream) {
  (void)n_in; (void)out_size; (void)ws_size;
  const float* img = (const float*)d_in[0];
  const float* txt = (const float*)d_in[1];
  const float* r   = (const float*)d_in[2];
  const float* alp = (const float*)d_in[3];
  const float* scl = (const float*)d_in[4];
  const float* lsc = (const float*)d_in[5];
  float* out = (float*)d_out;

  const int K = 16, D = 256;
  const int B = in_sizes[0] / (K * D);   // 4
  const int N = in_sizes[1] / (K * D);   // 2048

  float* wsGI = (float*)d_ws;            // B*256 floats
  float* wsS1 = wsGI + (size_t)B * 256;  // B*256 floats

  prep_img_kernel<<<B, 32, 0, stream>>>(img, r, wsGI, wsS1, (float)B);
  per_text_kernel<<<N, 32, 0, stream>>>(txt, r, alp, scl, wsGI, wsS1, out, N, (float)N);
  row_log_softmax_kernel<<<B, 256, 0, stream>>>(out, lsc, N, B);
}